// GAT_38766374814260
// MI455X (gfx1250) — compile-verified
//
#include <hip/hip_runtime.h>

typedef float v2f __attribute__((ext_vector_type(2)));
typedef float v8f __attribute__((ext_vector_type(8)));

#define NN 50000            // nodes
#define NE 800000           // edges
#define FDIM 256            // HEADS * C (both layers)
#define NH 4                // heads

__device__ __forceinline__ float selu_f(float x) {
  const float scale = 1.0507009873554805f;
  const float alpha = 1.6732632423543772f;
  return x > 0.f ? scale * x : scale * alpha * (__expf(x) - 1.f);
}

__device__ __forceinline__ float lrelu_f(float x) { return x > 0.f ? x : 0.2f * x; }

// ---- fills -----------------------------------------------------------------

__global__ void k_fill(float* __restrict__ p, float v, long long n) {
  long long i = (long long)blockIdx.x * blockDim.x + threadIdx.x;
  if (i < n) p[i] = v;
}

__global__ void k_fill4(float4* __restrict__ p, float v, long long n4) {
  long long i = (long long)blockIdx.x * blockDim.x + threadIdx.x;
  if (i < n4) p[i] = make_float4(v, v, v, v);
}

// ---- projection GEMM: Y[M,256] = X[M,K] @ W[K,256] via V_WMMA_F32_16X16X4_F32
// 512 threads = 16 waves; wave w owns N-tile w; grid.x = M/16 (M multiple of 16).
__global__ void k_gemm_wmma(const float* __restrict__ X, const float* __restrict__ W,
                            float* __restrict__ Y, int K) {
  extern __shared__ float Xs[];           // 16 x (K+4) padded tile of X
  const int stride = K + 4;
  const int tid  = threadIdx.x;
  const int wave = tid >> 5;
  const int lane = tid & 31;
  const int mBase = blockIdx.x * 16;

  for (int i = tid; i < 16 * K; i += blockDim.x) {
    int r = i / K, c = i - r * K;
    Xs[r * stride + c] = X[(long long)(mBase + r) * K + c];
  }
  __syncthreads();

  const int nBase = wave * 16;
  const int half  = lane >> 4;            // 0: K pair {0,1} / M rows 0-7 ; 1: {2,3} / rows 8-15
  const int l16   = lane & 15;
  v8f acc = {0.f, 0.f, 0.f, 0.f, 0.f, 0.f, 0.f, 0.f};

  for (int k0 = 0; k0 < K; k0 += 4) {
    const int ka = k0 + half * 2;
    v2f a, b;
    // A-matrix 16x4 f32: lane<16 rows M=lane hold K={0,1}; lane>=16 hold K={2,3}
    a.x = Xs[l16 * stride + ka];
    a.y = Xs[l16 * stride + ka + 1];
    // B-matrix 4x16: row-striped across lanes within a VGPR (mirrors A halves)
    b.x = W[(long long)ka * FDIM + nBase + l16];
    b.y = W[(long long)(ka + 1) * FDIM + nBase + l16];
    // Unconditional speculative prefetch of the next W k-slab (global_prefetch_b8).
    // Past-the-end addresses are speculative prefetches: dropped silently (ISA 10.5),
    // so no guard branch is needed and the WMMA clause stays straight-line.
    __builtin_prefetch(&W[(long long)(ka + 4) * FDIM + nBase + l16], 0, 1);
    acc = __builtin_amdgcn_wmma_f32_16x16x4_f32(false, a, false, b, (short)0, acc,
                                                false, false);
  }
  // C/D layout: VGPR v -> M = v + 8*half, N = l16
  for (int v = 0; v < 8; ++v) {
    int m = mBase + v + half * 8;
    Y[(long long)m * FDIM + nBase + l16] = acc[v];
  }
}

// ---- per-node attention logits: es[n,h]=<xp[n,h,:],a_self[h,:]>, en likewise.
__global__ void k_att(const float* __restrict__ xp, const float* __restrict__ a_s,
                      const float* __restrict__ a_n, float* __restrict__ es,
                      float* __restrict__ en, int N) {
  long long i = (long long)blockIdx.x * blockDim.x + threadIdx.x;
  if (i >= (long long)N * NH) return;
  int h = (int)(i & 3);
  const float* row = xp + (i >> 2) * FDIM + h * 64;
  const float* ws = a_s + h * 64;
  const float* wn = a_n + h * 64;
  float s = 0.f, t = 0.f;
  for (int c = 0; c < 64; ++c) { s += row[c] * ws[c]; t += row[c] * wn[c]; }
  es[i] = s; en[i] = t;
}

// ---- edge pass 1: e = leaky_relu(es[tgt]+en[src]); store; segment float-max.
// One thread per EDGE; heads handled as a float4 (rows are 16B aligned).
__global__ void k_edge_max(const long long* __restrict__ tgt, const long long* __restrict__ src,
                           const float* __restrict__ es, const float* __restrict__ en,
                           float* __restrict__ ebuf, float* __restrict__ m) {
  long long e = (long long)blockIdx.x * blockDim.x + threadIdx.x;
  if (e >= NE) return;
  long long t = tgt[e], s = src[e];
  float4 vs = *(const float4*)(es + t * NH);
  float4 vn = *(const float4*)(en + s * NH);
  float4 v;
  v.x = lrelu_f(vs.x + vn.x);
  v.y = lrelu_f(vs.y + vn.y);
  v.z = lrelu_f(vs.z + vn.z);
  v.w = lrelu_f(vs.w + vn.w);
  *(float4*)(ebuf + e * NH) = v;
  float* mt = m + t * NH;
  atomicMax(&mt[0], v.x);
  atomicMax(&mt[1], v.y);
  atomicMax(&mt[2], v.z);
  atomicMax(&mt[3], v.w);
}

// ---- edge pass 2: ex = exp(e - m[tgt]); store; segment sum of denom.
__global__ void k_edge_exp(const long long* __restrict__ tgt, float* __restrict__ ebuf,
                           const float* __restrict__ m, float* __restrict__ denom) {
  long long e = (long long)blockIdx.x * blockDim.x + threadIdx.x;
  if (e >= NE) return;
  long long t = tgt[e];
  float4 v  = *(const float4*)(ebuf + e * NH);
  float4 mv = *(const float4*)(m + t * NH);
  float4 ex;
  ex.x = __expf(v.x - mv.x);
  ex.y = __expf(v.y - mv.y);
  ex.z = __expf(v.z - mv.z);
  ex.w = __expf(v.w - mv.w);
  *(float4*)(ebuf + e * NH) = ex;
  float* dt = denom + t * NH;
  atomicAdd(&dt[0], ex.x);
  atomicAdd(&dt[1], ex.y);
  atomicAdd(&dt[2], ex.z);
  atomicAdd(&dt[3], ex.w);
}

// ---- edge pass 3: agg[tgt,c] += (ex/denom[tgt]) * xp[src,c]
// 64 threads per edge, 4 channels each (b128 gather); 4 edges per 256-thread block.
__global__ void k_agg(const long long* __restrict__ tgt, const long long* __restrict__ src,
                      const float* __restrict__ ebuf, const float* __restrict__ denom,
                      const float* __restrict__ xp, float* __restrict__ agg) {
  int sub = threadIdx.x >> 6;                       // edge slot within block (0..3)
  int t64 = threadIdx.x & 63;                       // 0..63
  long long e = (long long)blockIdx.x * 4 + sub;
  int c0 = t64 * 4;                                 // 4 consecutive channels
  int h  = c0 >> 6;
  long long t = tgt[e], s = src[e];
  float alpha = ebuf[e * NH + h] / denom[t * NH + h];
  float4 xv = *(const float4*)(xp + s * FDIM + c0);
  float* ao = agg + t * FDIM + c0;
  atomicAdd(&ao[0], alpha * xv.x);
  atomicAdd(&ao[1], alpha * xv.y);
  atomicAdd(&ao[2], alpha * xv.z);
  atomicAdd(&ao[3], alpha * xv.w);
}

// ---- bias + SELU, vectorized
__global__ void k_finalize(float4* __restrict__ agg, const float4* __restrict__ b, int N) {
  long long i = (long long)blockIdx.x * blockDim.x + threadIdx.x;  // over N*64 float4s
  if (i >= (long long)N * (FDIM / 4)) return;
  int c4 = (int)(i & (FDIM / 4 - 1));
  float4 v = agg[i], bb = b[c4];
  v.x = selu_f(v.x + bb.x);
  v.y = selu_f(v.y + bb.y);
  v.z = selu_f(v.z + bb.z);
  v.w = selu_f(v.w + bb.w);
  agg[i] = v;
}

// ---- GlobalSumPool: 64 threads/block, thread owns a float4 column group,
// block sums 128 rows, then 4 atomicAdds.
__global__ void k_pool(const float4* __restrict__ h, float* __restrict__ g, int N) {
  int c4 = threadIdx.x;                    // 0..63
  int r0 = blockIdx.x * 128;
  int r1 = r0 + 128; if (r1 > N) r1 = N;
  float4 s = make_float4(0.f, 0.f, 0.f, 0.f);
  for (int r = r0; r < r1; ++r) {
    float4 v = h[(long long)r * (FDIM / 4) + c4];
    s.x += v.x; s.y += v.y; s.z += v.z; s.w += v.w;
  }
  float* go = g + c4 * 4;
  atomicAdd(&go[0], s.x);
  atomicAdd(&go[1], s.y);
  atomicAdd(&go[2], s.z);
  atomicAdd(&go[3], s.w);
}

// ---- head MLP: 256 -> 64 -> 32 -> 1, single block of 64 threads.
__global__ void k_mlp(const float* __restrict__ g,
                      const float* __restrict__ Wd1, const float* __restrict__ bd1,
                      const float* __restrict__ Wd2, const float* __restrict__ bd2,
                      const float* __restrict__ Wo,  const float* __restrict__ bo,
                      float* __restrict__ out) {
  __shared__ float h1[64];
  __shared__ float h2[32];
  int t = threadIdx.x;                    // 64 threads
  float s = bd1[t];
  for (int k = 0; k < FDIM; ++k) s += g[k] * Wd1[k * 64 + t];
  h1[t] = selu_f(s);
  __syncthreads();
  if (t < 32) {
    float u = bd2[t];
    for (int k = 0; k < 64; ++k) u += h1[k] * Wd2[k * 32 + t];
    h2[t] = selu_f(u);
  }
  __syncthreads();
  if (t == 0) {
    float o = bo[0];
    for (int k = 0; k < 32; ++k) o += h2[k] * Wo[k];
    out[0] = o;
  }
}

extern "C" void kernel_launch(void* const* d_in, const int* in_sizes, int n_in,
                              void* d_out, int out_size, void* d_ws, size_t ws_size,
                              hipStream_t stream) {
  const float*     x   = (const float*)d_in[0];
  const long long* ei  = (const long long*)d_in[1];   // int64 [2, E]
  const float*     W1  = (const float*)d_in[2];
  const float*     as1 = (const float*)d_in[3];
  const float*     an1 = (const float*)d_in[4];
  const float*     b1  = (const float*)d_in[5];
  const float*     W2  = (const float*)d_in[6];
  const float*     as2 = (const float*)d_in[7];
  const float*     an2 = (const float*)d_in[8];
  const float*     b2  = (const float*)d_in[9];
  const float*     Wd1 = (const float*)d_in[10];
  const float*     bd1 = (const float*)d_in[11];
  const float*     Wd2 = (const float*)d_in[12];
  const float*     bd2 = (const float*)d_in[13];
  const float*     Wo  = (const float*)d_in[14];
  const float*     bo  = (const float*)d_in[15];
  float* out = (float*)d_out;

  const long long* tgt = ei;
  const long long* src = ei + NE;

  // workspace layout (floats; every region is 16B aligned)
  float* A   = (float*)d_ws;                        // xp (per layer)      [N,256]
  float* B   = A   + (size_t)NN * FDIM;             // layer1 agg/output   [N,256]
  float* D   = B   + (size_t)NN * FDIM;             // layer2 agg/output   [N,256]
  float* EB  = D   + (size_t)NN * FDIM;             // edge logits/ex      [E,4]
  float* ES  = EB  + (size_t)NE * NH;               // e_self              [N,4]
  float* EN  = ES  + (size_t)NN * NH;               // e_neigh             [N,4]
  float* MM  = EN  + (size_t)NN * NH;               // segment max         [N,4]
  float* DEN = MM  + (size_t)NN * NH;               // segment denom       [N,4]
  float* G   = DEN + (size_t)NN * NH;               // pooled              [256]

  const int THR = 256;
  const long long nNH  = (long long)NN * NH;
  const long long nNF4 = (long long)NN * (FDIM / 4);
  const int gNH  = (int)((nNH + THR - 1) / THR);
  const int gNF4 = (int)((nNF4 + THR - 1) / THR);
  const int gE   = (NE + THR - 1) / THR;
  const float NEG_INF = -__builtin_inff();

  // ---------------- layer 1 ----------------
  k_gemm_wmma<<<NN / 16, 512, (size_t)16 * (128 + 4) * 4, stream>>>(x, W1, A, 128);
  k_att<<<gNH, THR, 0, stream>>>(A, as1, an1, ES, EN, NN);
  k_fill<<<gNH, THR, 0, stream>>>(MM, NEG_INF, nNH);
  k_fill<<<gNH, THR, 0, stream>>>(DEN, 0.f, nNH);
  k_edge_max<<<gE, THR, 0, stream>>>(tgt, src, ES, EN, EB, MM);
  k_edge_exp<<<gE, THR, 0, stream>>>(tgt, EB, MM, DEN);
  k_fill4<<<gNF4, THR, 0, stream>>>((float4*)B, 0.f, nNF4);
  k_agg<<<NE / 4, THR, 0, stream>>>(tgt, src, EB, DEN, A, B);
  k_finalize<<<gNF4, THR, 0, stream>>>((float4*)B, (const float4*)b1, NN);

  // ---------------- layer 2 ----------------
  k_gemm_wmma<<<NN / 16, 512, (size_t)16 * (256 + 4) * 4, stream>>>(B, W2, A, 256);
  k_att<<<gNH, THR, 0, stream>>>(A, as2, an2, ES, EN, NN);
  k_fill<<<gNH, THR, 0, stream>>>(MM, NEG_INF, nNH);
  k_fill<<<gNH, THR, 0, stream>>>(DEN, 0.f, nNH);
  k_edge_max<<<gE, THR, 0, stream>>>(tgt, src, ES, EN, EB, MM);
  k_edge_exp<<<gE, THR, 0, stream>>>(tgt, EB, MM, DEN);
  k_fill4<<<gNF4, THR, 0, stream>>>((float4*)D, 0.f, nNF4);
  k_agg<<<NE / 4, THR, 0, stream>>>(tgt, src, EB, DEN, A, D);
  k_finalize<<<gNF4, THR, 0, stream>>>((float4*)D, (const float4*)b2, NN);

  // ---------------- pool + head MLP ----------------
  k_fill<<<1, FDIM, 0, stream>>>(G, 0.f, FDIM);
  k_pool<<<(NN + 127) / 128, 64, 0, stream>>>((const float4*)D, G, NN);
  k_mlp<<<1, 64, 0, stream>>>(G, Wd1, bd1, Wd2, bd2, Wo, bo, out);
}